// CTCLoss_51702816309284
// MI455X (gfx1250) — compile-verified
//
#include <hip/hip_runtime.h>
#include <math.h>

typedef __attribute__((ext_vector_type(2))) float v2f;
typedef __attribute__((ext_vector_type(8))) float v8f;
typedef __attribute__((ext_vector_type(4))) unsigned uint32x4;
typedef __attribute__((ext_vector_type(4))) int      int32x4;
typedef __attribute__((ext_vector_type(8))) int      int32x8;

#define NB    1024
#define NCLS  96
#define MT    256
#define ULAB  40
#define LST   81
#define LPAD  84   // 81 states padded to 21 K-chunks of 4 (pad cols are zero)

// LDS layout, in floats
#define OFF_P     0                 // probs tile: 96*256      = 24576
#define OFF_AP    24576             // A' = alpha/p : 256*84   = 21504
#define OFF_BC    46080             // B (time+state flipped)  = 21504
#define OFF_DIAG  67584             // per-wave 16x16 scratch: 8*256 = 2048
#define OFF_WP    69632             // per-wave loss partials: 8
#define SMEM_FLOATS 69640           // 278,560 bytes

#if __has_builtin(__builtin_amdgcn_tensor_load_to_lds) && \
    __has_builtin(__builtin_amdgcn_s_wait_tensorcnt)
#define USE_TDM 1
#else
#define USE_TDM 0
#endif

extern "C" __global__ void __launch_bounds__(256, 1)
ctc_block_kernel(const float* __restrict__ outp,
                 const int*   __restrict__ label,
                 const int*   __restrict__ upw,
                 float*       __restrict__ ws)
{
    extern __shared__ float smem[];
    float* P  = smem + OFF_P;
    float* Ap = smem + OFF_AP;
    float* Bc = smem + OFF_BC;
    float* DI = smem + OFF_DIAG;
    float* WP = smem + OFF_WP;

    const int b    = blockIdx.x;
    const int tid  = threadIdx.x;
    const int lane = tid & 31;
    const int wv   = tid >> 5;
    const int T    = upw[b] >> 2;           // w // 4, T in [128,256]
    const int* lab = label + b * ULAB;

    // ---- phase 0: stage probs slice into LDS (TDM DMA), zero A'/B ----------
#if USE_TDM
    if (wv == 0) {
        // Tensor DMA descriptor: 2-D tensor 96 rows x 256 cols of f32,
        // tile = whole tensor, contiguous (stride 256), into LDS at P.
        const unsigned long long ga =
            (unsigned long long)(const void*)(outp + (size_t)b * (NCLS * MT));
        const unsigned lds_addr = (unsigned)(size_t)(void*)P;  // LDS byte offset
        uint32x4 g0;
        g0[0] = 1u;                                  // count=1, user desc
        g0[1] = lds_addr;                            // lds_addr[31:0]
        g0[2] = (unsigned)(ga & 0xFFFFFFFFull);      // global_addr[31:0]
        g0[3] = (unsigned)((ga >> 32) & 0x01FFFFFFull) | 0x80000000u; // type=2
        int32x8 g1;
        g1[0] = 0x00020000;            // workgroup_mask=0, data_size=2 (4B)
        g1[1] = (int)(MT << 16);       // tensor_dim0[15:0]=256 (bits 63:48)
        g1[2] = (int)(NCLS << 16);     // tensor_dim0[31:16]=0, tensor_dim1=96
        g1[3] = (int)(MT << 16);       // tensor_dim1 hi=0, tile_dim0=256
        g1[4] = (int)NCLS;             // tile_dim1=96, tile_dim2=0
        g1[5] = (int)MT;               // tensor_dim0_stride[31:0]=256
        g1[6] = 0;                     // stride0 hi, tensor_dim1_stride lo
        g1[7] = 0;                     // tensor_dim1_stride hi
        int32x4 g2; g2[0] = 1; g2[1] = 1; g2[2] = 0; g2[3] = 0;  // dims2/3 = 1
        int32x4 g3; g3[0] = 0; g3[1] = 0x00010000; g3[2] = 0; g3[3] = 0; // dim4=1
        int32x8 g4;                    // extra group (6-arg toolchain): zero
        g4[0] = 0; g4[1] = 0; g4[2] = 0; g4[3] = 0;
        g4[4] = 0; g4[5] = 0; g4[6] = 0; g4[7] = 0;
        __builtin_amdgcn_tensor_load_to_lds(g0, g1, g2, g3, g4, 0);
    }
#else
    {
        const float4* gp = (const float4*)(outp + (size_t)b * (NCLS * MT));
        float4* P4 = (float4*)P;
        for (int i = tid; i < (NCLS * MT) / 4; i += 256) P4[i] = gp[i];
    }
#endif
    {   // zero A'/B (overlaps with the tensor DMA)
        float4 z4; z4.x = z4.y = z4.z = z4.w = 0.f;
        float4* AB4 = (float4*)Ap;          // Ap and Bc are contiguous
        for (int i = tid; i < (2 * MT * LPAD) / 4; i += 256) AB4[i] = z4;
    }
#if USE_TDM
    if (wv == 0) __builtin_amdgcn_s_wait_tensorcnt(0);
#endif
    __syncthreads();

    // ---- phase 1: alpha scan on wave 0, beta scan on wave 1 ----------------
    if (wv < 2) {
        const bool beta = (wv == 1);
        const bool act2 = (lane + 64 < LST);   // lane's third state valid?
        int cls[3]; int skp[3];
        #pragma unroll
        for (int j = 0; j < 3; ++j) {
            int s = lane + 32 * j; if (s >= LST) s = LST - 1;  // clamp (masked later)
            int c, k;
            if (s & 1) {
                int i = (s - 1) >> 1;
                if (!beta) {
                    c = lab[i];
                    k = (s != 1) && (lab[i] != lab[i - 1]);
                } else {
                    int li = lab[ULAB - 1 - i];
                    c = (NCLS - 1) - li;                        // class-flipped
                    k = (s != 1) && (li != lab[ULAB - i]);      // reversed labels
                }
            } else { c = beta ? (NCLS - 1) : 0; k = 0; }        // blank
            cls[j] = c; skp[j] = k;
        }
        const float* pr0 = P + cls[0] * MT;
        const float* pr1 = P + cls[1] * MT;
        const float* pr2 = P + cls[2] * MT;
        float* dst = beta ? Bc : Ap;

        // prev row initialized to delta at s=0: generic step then reproduces row0
        float v0 = (lane == 0) ? 1.f : 0.f, v1 = 0.f, v2 = 0.f;

        for (int t = 0; t < T; ++t) {
            int start = LST - 2 * (T - t); if (start < 0) start = 0;
            // neighbor gathers: prev[s-1], prev[s-2] via in-wave shuffles
            float a1 = __shfl_up(v0, 1), a2 = __shfl_up(v0, 2);
            float b1 = __shfl_up(v1, 1), b2 = __shfl_up(v1, 2);
            float c1 = __shfl_up(v2, 1), c2 = __shfl_up(v2, 2);
            float h031 = __shfl(v0, 31), h030 = __shfl(v0, 30);
            float h131 = __shfl(v1, 31), h130 = __shfl(v1, 30);
            float pm1_0 = lane ? a1 : 0.f;
            float pm2_0 = (lane >= 2) ? a2 : 0.f;
            float pm1_1 = lane ? b1 : h031;
            float pm2_1 = (lane >= 2) ? b2 : ((lane == 1) ? h031 : h030);
            float pm1_2 = lane ? c1 : h131;
            float pm2_2 = (lane >= 2) ? c2 : ((lane == 1) ? h131 : h130);

            float q0 = v0 + pm1_0 + (skp[0] ? pm2_0 : 0.f);
            float q1 = v1 + pm1_1 + (skp[1] ? pm2_1 : 0.f);
            float q2 = v2 + pm1_2 + (skp[2] ? pm2_2 : 0.f);
            if (lane < start)                 q0 = 0.f;
            if (lane + 32 < start)            q1 = 0.f;
            if (!act2 || (lane + 64 < start)) q2 = 0.f;

            float w0 = q0 * pr0[t];
            float w1 = q1 * pr1[t];
            float w2 = act2 ? q2 * pr2[t] : 0.f;

            float cs = w0 + w1 + w2;
            #pragma unroll
            for (int m = 16; m; m >>= 1) cs += __shfl_xor(cs, m);
            float rn = (cs > 0.f) ? (1.f / cs) : 1.f;

            if (!beta) {
                // store alpha/p = preval * rn  (division by probs folded in)
                dst[t * LPAD + lane]      = q0 * rn;
                dst[t * LPAD + lane + 32] = q1 * rn;
                if (act2) dst[t * LPAD + lane + 64] = q2 * rn;
            } else {
                // store b[t][s] = braw[T-1-t][80-s] directly in combined layout
                int tr = T - 1 - t;
                dst[tr * LPAD + (80 - lane)] = w0 * rn;
                dst[tr * LPAD + (48 - lane)] = w1 * rn;
                if (act2) dst[tr * LPAD + (16 - lane)] = w2 * rn;
            }
            v0 = w0 * rn; v1 = w1 * rn; v2 = act2 ? w2 * rn : 0.f;
        }
    }
    __syncthreads();

    // ---- phase 2: lh[t] = diag( A'[16x84] x Bc^T[84x16] ) via WMMA ---------
    float partial = 0.f;
    const int mn = lane & 15;           // row (M) for A / col (N) for B
    const int kb = (lane >> 4) * 2;     // K sub-offset per lane half
    float* dsc = DI + wv * 256;

    for (int tb = wv; tb < 16; tb += 8) {          // wave-uniform
        if (tb * 16 >= T) break;                   // wave-uniform (T>=128)
        const int t0 = tb * 16;
        const float* ar = Ap + (t0 + mn) * LPAD;
        const float* br = Bc + (t0 + mn) * LPAD;
        v8f acc = {0.f, 0.f, 0.f, 0.f, 0.f, 0.f, 0.f, 0.f};
        #pragma unroll
        for (int k0 = 0; k0 < LPAD; k0 += 4) {
            v2f av; av.x = ar[k0 + kb]; av.y = ar[k0 + kb + 1];
            v2f bv; bv.x = br[k0 + kb]; bv.y = br[k0 + kb + 1];
            acc = __builtin_amdgcn_wmma_f32_16x16x4_f32(
                false, av, false, bv, (short)0, acc, false, false);
        }
        // spill 16x16 result to per-wave LDS scratch, read back the diagonal
        #pragma unroll
        for (int r = 0; r < 8; ++r) {
            int row = (lane < 16) ? r : (r + 8);
            dsc[row * 16 + mn] = acc[r];
        }
        asm volatile("s_wait_dscnt 0" ::: "memory");
        float lh = (lane < 16) ? dsc[lane * 16 + lane] : 0.f;
        asm volatile("" ::: "memory");   // keep the read before next iter's stores
        float contrib = ((lane < 16) && (t0 + lane < T)) ? -__logf(lh) : 0.f;
        #pragma unroll
        for (int m = 16; m; m >>= 1) contrib += __shfl_xor(contrib, m);
        partial += contrib;
    }
    if (lane == 0) WP[wv] = partial;
    __syncthreads();
    if (tid == 0) {
        float s = 0.f;
        #pragma unroll
        for (int i = 0; i < 8; ++i) s += WP[i];
        ws[b] = s;
    }
}

// Deterministic fixed-order final reduction of the 1024 per-batch losses.
extern "C" __global__ void __launch_bounds__(256)
ctc_reduce_kernel(const float* __restrict__ ws, float* __restrict__ out)
{
    __shared__ float sm[256];
    int tid = threadIdx.x;
    float s = 0.f;
    for (int i = tid; i < NB; i += 256) s += ws[i];   // fixed order per thread
    sm[tid] = s;
    __syncthreads();
    for (int off = 128; off; off >>= 1) {
        if (tid < off) sm[tid] += sm[tid + off];
        __syncthreads();
    }
    if (tid == 0) out[0] = sm[0];
}

extern "C" void kernel_launch(void* const* d_in, const int* in_sizes, int n_in,
                              void* d_out, int out_size, void* d_ws, size_t ws_size,
                              hipStream_t stream)
{
    (void)in_sizes; (void)n_in; (void)out_size; (void)ws_size;
    const float* outp  = (const float*)d_in[0];   // (1024, 96, 256) f32
    const int*   label = (const int*)d_in[1];     // (1024, 40) i32
    const int*   upw   = (const int*)d_in[2];     // (1024,) i32
    float* loss = (float*)d_out;                  // scalar f32
    float* wsb  = (float*)d_ws;                   // >= 1024 floats scratch

    const size_t smem_bytes = (size_t)SMEM_FLOATS * sizeof(float);
    (void)hipFuncSetAttribute(reinterpret_cast<const void*>(&ctc_block_kernel),
                              hipFuncAttributeMaxDynamicSharedMemorySize,
                              (int)smem_bytes);

    ctc_block_kernel<<<NB, 256, smem_bytes, stream>>>(outp, label, upw, wsb);
    ctc_reduce_kernel<<<1, 256, 0, stream>>>(wsb, loss);
}